// Conv2dLocal_3075196584577
// MI455X (gfx1250) — compile-verified
//
#include <hip/hip_runtime.h>

typedef __attribute__((ext_vector_type(2))) float v2f;
typedef __attribute__((ext_vector_type(8))) float v8f;

// ---- CDNA5 async global->LDS copies (ASYNCcnt-tracked, bypass VGPRs) ------
__device__ __forceinline__ void async_g2l_b128(unsigned lds_off, const float* g) {
  asm volatile("global_load_async_to_lds_b128 %0, %1, off"
               :: "v"(lds_off), "v"((unsigned long long)(uintptr_t)g) : "memory");
}
__device__ __forceinline__ void async_g2l_b32(unsigned lds_off, const float* g) {
  asm volatile("global_load_async_to_lds_b32 %0, %1, off"
               :: "v"(lds_off), "v"((unsigned long long)(uintptr_t)g) : "memory");
}
__device__ __forceinline__ void wait_async0() {
  asm volatile("s_wait_asynccnt 0x0" ::: "memory");
}

// ---------------------------------------------------------------------------
// Pass 1: transpose x [B*C=4096][H*W=1024] -> xt [H*W=1024][B*C=4096]
// ---------------------------------------------------------------------------
__global__ __launch_bounds__(256) void xt_transpose(const float* __restrict__ x,
                                                    float* __restrict__ xt) {
  __shared__ float tile[32][33];
  const int hw0 = blockIdx.x * 32;
  const int bc0 = blockIdx.y * 32;
  const int tx = threadIdx.x, ty = threadIdx.y;
#pragma unroll
  for (int i = ty; i < 32; i += 8)
    tile[i][tx] = x[(size_t)(bc0 + i) * 1024 + hw0 + tx];
  __syncthreads();
#pragma unroll
  for (int i = ty; i < 32; i += 8)
    xt[(size_t)(hw0 + i) * 4096 + bc0 + tx] = tile[tx][i];
}

// ---------------------------------------------------------------------------
// Pass 2: one WG per output location (1024 WGs, 128 thr = 4 wave32).
// out[64b x 64o] = sum over 9 taps of A_tap[64x64] * B_tap[64x64], fp32 WMMA.
//   xt   : [hw][b*64 + c]
//   wgt  : [l][o][c][tap]
//   bias : [o][l]
//   out  : [b][o][l]
// ---------------------------------------------------------------------------
__global__ __launch_bounds__(128) void conv_local_wmma(
    const float* __restrict__ xt, const float* __restrict__ wgt,
    const float* __restrict__ bias, float* __restrict__ out) {
  // lA: [b][c], row stride 68 -> A-frag b64 loads hit all 64 banks.
  // lB: K-pair interleaved, (k,o) at (k>>1)*160 + o*2 + (k&1):
  //     a lane's B frag = 2 adjacent floats -> single ds_load_b64;
  //     row stride 160 (mod 64 banks = 32) keeps the two 16-lane halves
  //     on disjoint 32-bank windows.
  __shared__ __align__(16) float lA[64 * 68];   // 17.0 KB
  __shared__ __align__(16) float lB[32 * 160];  // 20.0 KB

  const int l = blockIdx.x;
  const int oh = l >> 5, ow = l & 31;
  const int tid = threadIdx.x;
  const int lane = tid & 31;
  const int wv = tid >> 5;
  const int m = lane & 15;
  const int h = lane >> 4;
  const int mBase = (wv >> 1) << 5;
  const int nBase = (wv & 1) << 5;

  v8f acc[2][2];
#pragma unroll
  for (int mt = 0; mt < 2; ++mt)
#pragma unroll
    for (int nt = 0; nt < 2; ++nt) {
      const float bv = bias[(size_t)(nBase + nt * 16 + m) * 1024 + l];
      acc[mt][nt] = (v8f){bv, bv, bv, bv, bv, bv, bv, bv};
    }

  const float* __restrict__ wl = wgt + (size_t)l * (64 * 64 * 9);

  // ---- tap-invariant per-thread staging geometry ----
  // A: thread covers float4 f = tid + 128*i  ->  b = tid>>4 (+8/i), c4 = tid&15
  const int a_c4 = tid & 15, a_b0 = tid >> 4;
  const unsigned lA_t = (unsigned)(uintptr_t)&lA[0] +
                        (unsigned)((a_b0 * 68 + a_c4 * 4) * 4);
  // B: element e = tid + 128*i -> o = (tid>>6) + 2*i, c = tid&63
  const int b_c = tid & 63, b_o0 = tid >> 6;
  const unsigned lB_t = (unsigned)(uintptr_t)&lB[0] +
                        (unsigned)((((b_c >> 1) * 160) + b_o0 * 2 + (b_c & 1)) * 4);
  const float* wt_t = wl + b_o0 * 576 + b_c * 9;

  for (int tap = 0; tap < 9; ++tap) {
    const int kh = tap / 3, kw = tap - kh * 3;
    const int ih = oh + kh - 1, iw = ow + kw - 1;
    const bool inb = ((unsigned)ih < 32u) & ((unsigned)iw < 32u);

    // ---- stage A_tap[b][c]: contiguous 16 KB, async b128 direct-to-LDS ----
    if (inb) {
      const float* ga = xt + ((size_t)((ih << 5) + iw) << 12) + tid * 4;
#pragma unroll
      for (int i = 0; i < 8; ++i)
        async_g2l_b128(lA_t + (unsigned)(i * 2176), ga + i * 512);
    } else {
#pragma unroll
      for (int i = 0; i < 8; ++i)
        *(float4*)(&lA[(a_b0 + 8 * i) * 68 + a_c4 * 4]) =
            make_float4(0.f, 0.f, 0.f, 0.f);
    }
    // ---- stage B_tap: per-lane stride-9 gather, async b32 direct-to-LDS ----
    {
      const float* gw = wt_t + tap;
#pragma unroll
      for (int i = 0; i < 32; ++i)
        async_g2l_b32(lB_t + (unsigned)(i * 16), gw + i * 1152);
    }
    wait_async0();
    __syncthreads();

#pragma unroll
    for (int k0 = 0; k0 < 64; k0 += 4) {
      v2f a[2], bf[2];
      // A frag (16x4): lanes 0-15 -> K={k0,k0+1}, lanes 16-31 -> K={k0+2,k0+3}
#pragma unroll
      for (int mt = 0; mt < 2; ++mt) {
        const float* ap = &lA[(mBase + mt * 16 + m) * 68 + k0 + 2 * h];
        a[mt] = (v2f){ap[0], ap[1]};
      }
      // B frag: pair-interleaved -> one aligned b64 per lane
#pragma unroll
      for (int nt = 0; nt < 2; ++nt) {
        const float* bp = &lB[((k0 >> 1) + h) * 160 + (nBase + nt * 16 + m) * 2];
        bf[nt] = (v2f){bp[0], bp[1]};
      }
#pragma unroll
      for (int mt = 0; mt < 2; ++mt)
#pragma unroll
        for (int nt = 0; nt < 2; ++nt)
          acc[mt][nt] = __builtin_amdgcn_wmma_f32_16x16x4_f32(
              false, a[mt], false, bf[nt], (short)0, acc[mt][nt], false, false);
    }
    __syncthreads();
  }

  // epilogue: C layout VGPR r -> M = r + 8*h, N = m
#pragma unroll
  for (int mt = 0; mt < 2; ++mt)
#pragma unroll
    for (int nt = 0; nt < 2; ++nt) {
      const int o = nBase + nt * 16 + m;
#pragma unroll
      for (int r = 0; r < 8; ++r) {
        const int bb = mBase + mt * 16 + r + (h << 3);
        out[((size_t)bb << 16) + (size_t)o * 1024 + l] = acc[mt][nt][r];
      }
    }
}

extern "C" void kernel_launch(void* const* d_in, const int* in_sizes, int n_in,
                              void* d_out, int out_size, void* d_ws, size_t ws_size,
                              hipStream_t stream) {
  const float* x    = (const float*)d_in[0];  // [64,64,32,32]
  const float* w    = (const float*)d_in[1];  // [32,32,64,64,3,3]
  const float* bias = (const float*)d_in[2];  // [64,32,32]
  float* out = (float*)d_out;                 // [64,64,32,32]
  float* xt  = (float*)d_ws;                  // 16 MiB scratch

  dim3 tb(32, 8);
  dim3 tg(1024 / 32, 4096 / 32);
  xt_transpose<<<tg, tb, 0, stream>>>(x, xt);
  conv_local_wmma<<<1024, 128, 0, stream>>>(xt, w, bias, out);
}